// RefGate_2911987827144
// MI455X (gfx1250) — compile-verified
//
#include <hip/hip_runtime.h>
#include <math.h>
#include <stdint.h>

#define DIM 7168
#define NEXP 256
#define NGROUPS 8
#define GSIZE 32
#define TOPKG 4
#define TOPK 8
#define ROUTE_SCALE 2.5f

#define MROWS 32                                // rows per block (2 x 16-row WMMA tiles)
#define KT 32                                   // K-tile depth
#define ASTRIDE 36                              // KT + 4 pad (16B-aligned rows, conflict-free frags)
#define TILE_FLOATS ((MROWS + NEXP) * ASTRIDE)  // 10368 floats per buffer
#define SSTRIDE 258                             // score overlay stride (32*258=8256 <= 10368)

typedef __attribute__((ext_vector_type(2))) float v2f;
typedef __attribute__((ext_vector_type(8))) float v8f;

// ---- CDNA5 async global->LDS (inline asm: portable across toolchains) ----
__device__ __forceinline__ void async_b128(unsigned ldsByte, const float* g) {
    asm volatile("global_load_async_to_lds_b128 %0, %1, off"
                 :: "v"(ldsByte), "v"((unsigned long long)(uintptr_t)g)
                 : "memory");
}
__device__ __forceinline__ void wait_async0() {
    asm volatile("s_wait_asynccnt 0" ::: "memory");
}

// One K-tile: A (32x32 f32) = 1x b128/thread, B (256x32 f32) = 8x b128/thread.
// Uniform 9 async ops per thread -> per-wave s_wait_asynccnt 0 is an exact handshake.
__device__ __forceinline__ void issue_tile(const float* __restrict__ x,
                                           const float* __restrict__ w,
                                           int row0, int k0,
                                           unsigned ldsBufByte, int t) {
    const int c4 = (t & 7) * 4;
    {   // A tile: rows 0..31
        const int m = t >> 3;
        async_b128(ldsBufByte + (unsigned)(m * ASTRIDE + c4) * 4u,
                   x + (size_t)(row0 + m) * DIM + k0 + c4);
    }
    {   // B tile (expert-major rows)
        const int e0 = t >> 3;
        const float* g = w + (size_t)e0 * DIM + k0 + c4;
        unsigned l = ldsBufByte + (unsigned)((MROWS + e0) * ASTRIDE + c4) * 4u;
#pragma unroll
        for (int i = 0; i < 8; ++i) {
            async_b128(l, g);
            g += (size_t)32 * DIM;
            l += 32u * ASTRIDE * 4u;
        }
    }
}

__global__ __launch_bounds__(256) void moe_gate_kernel(
    const float* __restrict__ x, const float* __restrict__ w,
    const float* __restrict__ bias, float* __restrict__ outw,
    float* __restrict__ outi)
{
    extern __shared__ float lds[];              // [2 * TILE_FLOATS] tiles + [NEXP] bias
    float* biasS = lds + 2 * TILE_FLOATS;

    const int t    = threadIdx.x;
    const int wave = t >> 5;
    const int lane = t & 31;
    const int half = lane >> 4;
    const int lr   = lane & 15;
    const int row0 = blockIdx.x * MROWS;

    const unsigned ldsBase  = (unsigned)(uintptr_t)lds;  // flat shared addr low bits = LDS offset
    const unsigned bufByte0 = ldsBase;
    const unsigned bufByte1 = ldsBase + (unsigned)TILE_FLOATS * 4u;

    if (t < NEXP) biasS[t] = bias[t];

    v8f acc00 = {}, acc01 = {};   // rows 0-15  x experts {32w..+15, +16..+31}
    v8f acc10 = {}, acc11 = {};   // rows 16-31 x same experts

    // prologue: fill buffer 0 with tile 0
    issue_tile(x, w, row0, 0, bufByte0, t);
    wait_async0();
    __syncthreads();

    const int NT = DIM / KT;  // 224
    for (int kt = 0; kt < NT; ++kt) {
        const int cur = kt & 1;
        // kick next tile's async fills into the other buffer; they overlap the WMMAs below
        if (kt + 1 < NT)
            issue_tile(x, w, row0, (kt + 1) * KT, cur ? bufByte0 : bufByte1, t);

        const float* As = lds + cur * TILE_FLOATS;
        const float* Bs = As + MROWS * ASTRIDE;
        // A frag (16x4 f32): lanes 0-15 -> row lr, cols k,k+1 ; lanes 16-31 -> cols k+2,k+3
        const float* arow0 = As + lr * ASTRIDE + 2 * half;           // rows 0-15
        const float* arow1 = arow0 + 16 * ASTRIDE;                   // rows 16-31
        const float* brow0 = Bs + (wave * 32 + lr) * ASTRIDE + 2 * half;
        const float* brow1 = brow0 + 16 * ASTRIDE;
#pragma unroll
        for (int kk = 0; kk < KT; kk += 4) {
            v2f a0 = *(const v2f*)(arow0 + kk);
            v2f a1 = *(const v2f*)(arow1 + kk);
            v2f b0 = *(const v2f*)(brow0 + kk);
            v2f b1 = *(const v2f*)(brow1 + kk);
            acc00 = __builtin_amdgcn_wmma_f32_16x16x4_f32(false, a0, false, b0,
                                                          (short)0, acc00, false, false);
            acc01 = __builtin_amdgcn_wmma_f32_16x16x4_f32(false, a0, false, b1,
                                                          (short)0, acc01, false, false);
            acc10 = __builtin_amdgcn_wmma_f32_16x16x4_f32(false, a1, false, b0,
                                                          (short)0, acc10, false, false);
            acc11 = __builtin_amdgcn_wmma_f32_16x16x4_f32(false, a1, false, b1,
                                                          (short)0, acc11, false, false);
        }
        wait_async0();       // own share of tile kt+1 landed
        __syncthreads();     // everyone's reads of buf[cur] + async fills done
    }

    // ---- sigmoid + write scores S[MROWS][SSTRIDE] (overlays buffer 0) ----
    float* S = lds;
    {
        const int colBase = wave * 32 + lr;
#pragma unroll
        for (int r = 0; r < 8; ++r) {
            const int m0 = half ? (r + 8) : r;    // C/D layout: vgpr r = row r / r+8
            const int m1 = m0 + 16;
            S[m0 * SSTRIDE + colBase]      = 1.f / (1.f + __expf(-acc00[r]));
            S[m0 * SSTRIDE + colBase + 16] = 1.f / (1.f + __expf(-acc01[r]));
            S[m1 * SSTRIDE + colBase]      = 1.f / (1.f + __expf(-acc10[r]));
            S[m1 * SSTRIDE + colBase + 16] = 1.f / (1.f + __expf(-acc11[r]));
        }
    }
    __syncthreads();

    // ---- per-row routing: MROWS threads, one row each ----
    if (t < MROWS) {
        const float* sr = S + t * SSTRIDE;

        float gs[NGROUPS];
        for (int g = 0; g < NGROUPS; ++g) {
            float m1 = -INFINITY, m2 = -INFINITY;
            for (int e = g * GSIZE; e < (g + 1) * GSIZE; ++e) {
                float v = sr[e] + biasS[e];
                if (v > m1)      { m2 = m1; m1 = v; }
                else if (v > m2) { m2 = v; }
            }
            gs[g] = m1 + m2;
        }
        bool keep[NGROUPS];
        for (int g = 0; g < NGROUPS; ++g) keep[g] = false;
        for (int j = 0; j < TOPKG; ++j) {
            int best = 0; float bv = -INFINITY;
            for (int g = 0; g < NGROUPS; ++g)
                if (!keep[g] && gs[g] > bv) { bv = gs[g]; best = g; }
            keep[best] = true;
        }
        int   idx[TOPK];
        float wsel[TOPK];
        unsigned taken[8] = {0, 0, 0, 0, 0, 0, 0, 0};
        for (int j = 0; j < TOPK; ++j) {
            int best = 0; float bv = -INFINITY;
            for (int e = 0; e < NEXP; ++e) {
                if (!keep[e >> 5]) continue;
                if (taken[e >> 5] & (1u << (e & 31))) continue;
                float v = sr[e] + biasS[e];
                if (v > bv) { bv = v; best = e; }
            }
            taken[best >> 5] |= (1u << (best & 31));
            idx[j]  = best;
            wsel[j] = sr[best];           // weights come from ORIGINAL sigmoid scores
        }
        float sum = 0.f;
        for (int j = 0; j < TOPK; ++j) sum += wsel[j];
        const float scale = ROUTE_SCALE / sum;
        const int grow = row0 + t;
        for (int j = 0; j < TOPK; ++j) {
            outw[(size_t)grow * TOPK + j] = wsel[j] * scale;
            outi[(size_t)grow * TOPK + j] = (float)idx[j];
        }
    }
}

extern "C" void kernel_launch(void* const* d_in, const int* in_sizes, int n_in,
                              void* d_out, int out_size, void* d_ws, size_t ws_size,
                              hipStream_t stream) {
    (void)n_in; (void)out_size; (void)d_ws; (void)ws_size;
    const float* x = (const float*)d_in[0];
    const float* w = (const float*)d_in[1];
    const float* b = (const float*)d_in[2];
    const int T = in_sizes[0] / DIM;      // 16384
    float* outw = (float*)d_out;
    float* outi = outw + (size_t)T * TOPK;
    const size_t shmem = (size_t)(2 * TILE_FLOATS + NEXP) * sizeof(float); // ~83 KB (320 KB/WGP)
    moe_gate_kernel<<<dim3(T / MROWS), dim3(256), shmem, stream>>>(x, w, b, outw, outi);
}